// NeighborhoodAttention2D_32203664785485
// MI455X (gfx1250) — compile-verified
//
#include <hip/hip_runtime.h>
#include <hip/hip_bf16.h>

typedef __attribute__((ext_vector_type(2))) float v2f;
typedef __attribute__((ext_vector_type(8))) float v8f;

// ---------------------------------------------------------------------------
// GEMM with bias:  C[M,N] = A[M,K] * W[N,K]^T + bias[N]
// One wave32 computes one 16x16 tile of C via V_WMMA_F32_16X16X4_F32 chained
// over K (K multiple of 4). A, W row-major with leading dims K; C leading N.
// ---------------------------------------------------------------------------
__global__ __launch_bounds__(256)
void gemm_bias_wmma_f32(const float* __restrict__ A,
                        const float* __restrict__ W,
                        const float* __restrict__ bias,
                        float* __restrict__ C,
                        int M, int N, int K)
{
    const int lane        = threadIdx.x & 31;
    const int waveInBlock = threadIdx.x >> 5;
    const int wavesPerBlk = blockDim.x >> 5;
    const int tilesN      = N >> 4;

    const int globalWave = blockIdx.x * wavesPerBlk + waveInBlock;
    const int tileM = globalWave / tilesN;
    const int tileN = globalWave % tilesN;
    if (tileM * 16 >= M) return;              // wave-uniform: EXEC stays all-1s

    const int laneLo = lane & 15;             // row within A-tile / col within C-tile
    const int kHalf  = (lane >> 4) << 1;      // 0 for lanes 0-15, 2 for lanes 16-31

    const int rowA = tileM * 16 + laneLo;     // A-matrix row for this lane
    const int rowW = tileN * 16 + laneLo;     // weight row == output column

    // C/D layout: VGPR r -> M = r + 8*(lane>>4); N = laneLo. Bias is per-column.
    const float bval = bias[tileN * 16 + laneLo];
    v8f c;
#pragma unroll
    for (int i = 0; i < 8; ++i) c[i] = bval;

    const float* aPtr = A + (size_t)rowA * K + kHalf;
    const float* wPtr = W + (size_t)rowW * K + kHalf;

#pragma unroll 8
    for (int k0 = 0; k0 < K; k0 += 4) {
        v2f a = *reinterpret_cast<const v2f*>(aPtr + k0);   // A[row][k0+kHalf .. +1]
        v2f b = *reinterpret_cast<const v2f*>(wPtr + k0);   // W[col][k0+kHalf .. +1]
        c = __builtin_amdgcn_wmma_f32_16x16x4_f32(
                /*neg_a=*/false, a, /*neg_b=*/false, b,
                /*c_mod=*/(short)0, c, /*reuse_a=*/false, /*reuse_b=*/false);
    }

    const int colC  = tileN * 16 + laneLo;
    const int mBase = tileM * 16 + ((lane >> 4) << 3);
#pragma unroll
    for (int r = 0; r < 8; ++r) {
        C[(size_t)(mBase + r) * N + colC] = c[r];
    }
}

// ---------------------------------------------------------------------------
// Fused 7x7 neighborhood attention (per-head). One thread = one (b,head,x,y).
// qkv layout: (N=B*H*W, 384) rows = [q(128) | k(128) | v(128)], head-major
// within each 128 (head*32 + d). Output: (N, 128) fp32.
// ---------------------------------------------------------------------------
__global__ __launch_bounds__(256)
void natten2d_k7(const float* __restrict__ qkv, float* __restrict__ out)
{
    constexpr int Hd = 64, Wd = 64, HD = 32, K3 = 384;
    const float scale = 0.17677669529663689f;   // 32^-0.5

    const int idx  = blockIdx.x * blockDim.x + threadIdx.x;
    const int y    = idx & 63;
    const int x    = (idx >> 6) & 63;
    const int head = (idx >> 12) & 3;
    const int b    = idx >> 14;

    const int n = ((b * Hd + x) << 6) + y;
    const float* qp = qkv + (size_t)n * K3 + head * HD;

    float q[HD];
#pragma unroll
    for (int d = 0; d < HD; ++d) q[d] = qp[d] * scale;

    int sh = x - 3; sh = sh < 0 ? 0 : sh; sh = sh > Hd - 7 ? Hd - 7 : sh;
    int sw = y - 3; sw = sw < 0 ? 0 : sw; sw = sw > Wd - 7 ? Wd - 7 : sw;

    float s[49];
    float mval = -3.402823466e38f;
#pragma unroll
    for (int i = 0; i < 7; ++i) {
        const float* kr = qkv + (size_t)(((b * Hd + sh + i) << 6) + sw) * K3 + 128 + head * HD;
#pragma unroll
        for (int j = 0; j < 7; ++j) {
            const float* kp = kr + (size_t)j * K3;
            float acc = 0.f;
#pragma unroll
            for (int d = 0; d < HD; ++d) acc = fmaf(q[d], kp[d], acc);
            s[i * 7 + j] = acc;
            mval = fmaxf(mval, acc);
        }
    }

    float sum = 0.f;
#pragma unroll
    for (int t = 0; t < 49; ++t) { s[t] = __expf(s[t] - mval); sum += s[t]; }
    const float inv = 1.f / sum;
#pragma unroll
    for (int t = 0; t < 49; ++t) s[t] *= inv;

    float o[HD];
#pragma unroll
    for (int d = 0; d < HD; ++d) o[d] = 0.f;
#pragma unroll
    for (int i = 0; i < 7; ++i) {
        const float* vr = qkv + (size_t)(((b * Hd + sh + i) << 6) + sw) * K3 + 256 + head * HD;
#pragma unroll
        for (int j = 0; j < 7; ++j) {
            const float* vp = vr + (size_t)j * K3;
            const float w = s[i * 7 + j];
#pragma unroll
            for (int d = 0; d < HD; ++d) o[d] = fmaf(w, vp[d], o[d]);
        }
    }

    float* op = out + (size_t)n * 128 + head * HD;
#pragma unroll
    for (int d = 0; d < HD; ++d) op[d] = o[d];
}

// ---------------------------------------------------------------------------
// Launch: qkv GEMM -> fused attention -> projection GEMM
// ---------------------------------------------------------------------------
extern "C" void kernel_launch(void* const* d_in, const int* in_sizes, int n_in,
                              void* d_out, int out_size, void* d_ws, size_t ws_size,
                              hipStream_t stream)
{
    const float* x      = (const float*)d_in[0];   // (8,64,64,128)
    const float* w_qkv  = (const float*)d_in[1];   // (384,128)
    const float* b_qkv  = (const float*)d_in[2];   // (384,)
    const float* w_proj = (const float*)d_in[3];   // (128,128)
    const float* b_proj = (const float*)d_in[4];   // (128,)
    float* out = (float*)d_out;                    // (8,64,64,128) fp32

    constexpr int Npix = 8 * 64 * 64;              // 32768 rows
    float* qkv  = (float*)d_ws;                            // 32768*384 f32 = 48 MB
    float* attn = qkv + (size_t)Npix * 384;                // 32768*128 f32 = 16 MB

    // 1) qkv = x @ w_qkv^T + b_qkv   (M=32768, N=384, K=128)
    {
        const int waves  = (Npix / 16) * (384 / 16);       // 49152 tiles
        const int blocks = waves / 8;                      // 8 waves/block
        gemm_bias_wmma_f32<<<blocks, 256, 0, stream>>>(x, w_qkv, b_qkv, qkv,
                                                       Npix, 384, 128);
    }

    // 2) fused neighborhood attention (per pixel, per head)
    {
        const int threads = 8 * 4 * 64 * 64;               // 131072
        natten2d_k7<<<threads / 256, 256, 0, stream>>>(qkv, attn);
    }

    // 3) out = attn @ w_proj^T + b_proj   (M=32768, N=128, K=128)
    {
        const int waves  = (Npix / 16) * (128 / 16);       // 16384 tiles
        const int blocks = waves / 8;
        gemm_bias_wmma_f32<<<blocks, 256, 0, stream>>>(attn, w_proj, b_proj, out,
                                                       Npix, 128, 128);
    }
}